// GPN_GCN_with_CRF_59442347377117
// MI455X (gfx1250) — compile-verified
//
#include <hip/hip_runtime.h>
#include <math.h>

// ---------------------------------------------------------------------------
// GPN GCN + CRF + APPNP pipeline for MI455X (gfx1250, wave32, WMMA f32)
// ---------------------------------------------------------------------------

#define N_NODES 50000
#define N_EDGES 800000
#define FEAT    128
#define HID     128
#define NCLS    40

typedef float v2f __attribute__((ext_vector_type(2)));
typedef float v8f __attribute__((ext_vector_type(8)));

__device__ __forceinline__ void atomAddF(float* p, float v) {
    // lowers to global_atomic_add_f32 (no return) on gfx1250
    __hip_atomic_fetch_add(p, v, __ATOMIC_RELAXED, __HIP_MEMORY_SCOPE_AGENT);
}

// ---------------------------------------------------------------- utilities
__global__ void k_fill(float* __restrict__ p, float v, long n) {
    long i = blockIdx.x * (long)blockDim.x + threadIdx.x;
    if (i < n) p[i] = v;
}

__global__ void k_deg_accum(const int* __restrict__ dst, float* __restrict__ deg, int E) {
    int e = blockIdx.x * blockDim.x + threadIdx.x;
    if (e < E) atomAddF(&deg[dst[e]], 1.0f);
}

__global__ void k_dinv(const float* __restrict__ deg, float* __restrict__ dinv, int n) {
    int i = blockIdx.x * blockDim.x + threadIdx.x;
    if (i < n) {
        float d = deg[i];
        dinv[i] = (d > 0.0f) ? rsqrtf(d) : 0.0f;
    }
}

// -------------------------------------------------------------- WMMA GEMM
// C[M,Nc] = A[M,K] * B[K,Nc] (+bias) (+relu), row-major, fp32 WMMA 16x16x4.
// One wave computes one 16x16 tile; block = 128 threads = 4 waves.
// K, Nc are compile-time: full K unroll, immediate-offset loads, and the
// Nc%16==0 path carries zero masking (no exec manipulation in the MMA loop).
template<int K, int Nc>
__global__ __launch_bounds__(128) void k_gemm_wmma(
    const float* __restrict__ A, const float* __restrict__ B,
    const float* __restrict__ bias, float* __restrict__ C,
    int M, int doRelu)
{
    constexpr int NcPad = (Nc + 15) & ~15;
    constexpr bool PARTIAL = (Nc & 15) != 0;

    const int lane = threadIdx.x & 31;
    const int wv   = threadIdx.x >> 5;
    const int row0 = blockIdx.x * 16;
    const int col0 = (blockIdx.y * 4 + wv) * 16;
    if (col0 >= NcPad) return;              // wave-uniform: EXEC stays full

    const int mIdx = lane & 15;             // row within tile (A), col (B/C)
    const int kHi  = lane >> 4;             // K half select
    const int col  = col0 + mIdx;

    // Branchless handling of the ragged last column tile: clamp the address
    // (stays in-bounds) and scale the fragment by 0 for dead columns.
    float bmask = 1.0f;
    int   colLd = col;
    if (PARTIAL) {
        if (col >= Nc) { bmask = 0.0f; colLd = Nc - 1; }
    }

    const float* ap = A + (size_t)(row0 + mIdx) * K + 2 * kHi;  // A[m][2*kHi]
    const float* bp = B + (size_t)(2 * kHi) * Nc + colLd;       // B[2*kHi][col]

    v8f acc = {};
#pragma unroll
    for (int kk = 0; kk < K; kk += 4) {
        v2f a, b;
        a.x = ap[0];                        // contiguous b64 load
        a.y = ap[1];
        b.x = bp[0];
        b.y = bp[Nc];
        if (PARTIAL) { b.x *= bmask; b.y *= bmask; }
        acc = __builtin_amdgcn_wmma_f32_16x16x4_f32(
            /*neg_a=*/false, a, /*neg_b=*/false, b,
            /*c_mod=*/(short)0, acc, /*reuse_a=*/false, /*reuse_b=*/false);
        ap += 4;
        bp += 4 * Nc;
    }

    if (!PARTIAL || col < Nc) {
        const float bv = bias ? bias[col] : 0.0f;
#pragma unroll
        for (int v = 0; v < 8; ++v) {
            int r = row0 + v + 8 * kHi;     // C layout: VGPR v, lanes16-31 = M+8
            if (r < M) {
                float val = acc[v] + bv;
                if (doRelu) val = fmaxf(val, 0.0f);
                C[(size_t)r * Nc + col] = val;
            }
        }
    }
}

// ------------------------------------------------- propagation (scatter-add)
// Self-loop term folded into init: out[n][:] = dinv[n]^2 * hin[n][:]
template<int W>   // W = floats per row
__global__ void k_init_self(const float* __restrict__ hin,
                            const float* __restrict__ dinv,
                            float* __restrict__ out, int n)
{
    long i = blockIdx.x * (long)blockDim.x + threadIdx.x;
    if (i < (long)n * W) {
        int node = (int)(i / W);
        float w = dinv[node];
        out[i] = (w * w) * hin[i];
    }
}

// out[dst] += dinv[src]*dinv[dst] * h[src], float4 chunks, W4 chunks per row
template<int W4>
__global__ void k_scatter(const int* __restrict__ src, const int* __restrict__ dst,
                          const float* __restrict__ dinv,
                          const float* __restrict__ hin, float* __restrict__ out,
                          int E)
{
    long t = blockIdx.x * (long)blockDim.x + threadIdx.x;
    if (t >= (long)E * W4) return;
    int e = (int)(t / W4);
    int c = (int)(t % W4);
    int s = src[e], d = dst[e];
    float w = dinv[s] * dinv[d];
    const float4 v = *((const float4*)(hin + (size_t)s * (W4 * 4)) + c);
    float* op = out + (size_t)d * (W4 * 4) + (size_t)c * 4;
    atomAddF(op + 0, w * v.x);
    atomAddF(op + 1, w * v.y);
    atomAddF(op + 2, w * v.z);
    atomAddF(op + 3, w * v.w);
}

// CRF: crf[src] += h[dst]; crf[dst] += h[src]   (BETA = 1)
template<int W4>
__global__ void k_crf_scatter(const int* __restrict__ src, const int* __restrict__ dst,
                              const float* __restrict__ h, float* __restrict__ crf,
                              int E)
{
    long t = blockIdx.x * (long)blockDim.x + threadIdx.x;
    if (t >= (long)E * W4) return;
    int e = (int)(t / W4);
    int c = (int)(t % W4);
    int s = src[e], d = dst[e];
    const float4 hs = *((const float4*)(h + (size_t)s * (W4 * 4)) + c);
    const float4 hd = *((const float4*)(h + (size_t)d * (W4 * 4)) + c);
    float* ps = crf + (size_t)s * (W4 * 4) + (size_t)c * 4;
    float* pd = crf + (size_t)d * (W4 * 4) + (size_t)c * 4;
    atomAddF(ps + 0, hd.x); atomAddF(ps + 1, hd.y);
    atomAddF(ps + 2, hd.z); atomAddF(ps + 3, hd.w);
    atomAddF(pd + 0, hs.x); atomAddF(pd + 1, hs.y);
    atomAddF(pd + 2, hs.z); atomAddF(pd + 3, hs.w);
}

// --------------------------------------------------------- elementwise ops
// float4 over 128-wide rows: chunk i covers cols 4*(i&31)..4*(i&31)+3
__global__ void k_bias_relu128(const float4* __restrict__ in, const float4* __restrict__ bias,
                               float4* __restrict__ out, long total4)
{
    long i = blockIdx.x * (long)blockDim.x + threadIdx.x;
    if (i < total4) {
        float4 v = in[i];
        float4 b = bias[i & 31];
        v.x = fmaxf(v.x + b.x, 0.0f);
        v.y = fmaxf(v.y + b.y, 0.0f);
        v.z = fmaxf(v.z + b.z, 0.0f);
        v.w = fmaxf(v.w + b.w, 0.0f);
        out[i] = v;
    }
}

__global__ void k_lerp(const float4* __restrict__ a, const float4* __restrict__ b,
                       float4* __restrict__ out, float ca, float cb, long total4)
{
    long i = blockIdx.x * (long)blockDim.x + threadIdx.x;
    if (i < total4) {
        float4 va = a[i], vb = b[i], r;
        r.x = ca * va.x + cb * vb.x;
        r.y = ca * va.y + cb * vb.y;
        r.z = ca * va.z + cb * vb.z;
        r.w = ca * va.w + cb * vb.w;
        out[i] = r;
    }
}

__global__ void k_log_softmax(const float* __restrict__ in, float* __restrict__ out, int n)
{
    int i = blockIdx.x * blockDim.x + threadIdx.x;
    if (i >= n) return;
    const float* p = in + (size_t)i * NCLS;
    float m = -__builtin_inff();
#pragma unroll
    for (int c = 0; c < NCLS; ++c) m = fmaxf(m, p[c]);
    float s = 0.0f;
#pragma unroll
    for (int c = 0; c < NCLS; ++c) s += expf(p[c] - m);
    float l = m + logf(s);
    float* o = out + (size_t)i * NCLS;
#pragma unroll
    for (int c = 0; c < NCLS; ++c) o[c] = p[c] - l;
}

// ---------------------------------------------------------------------------
static inline int cdiv(long a, long b) { return (int)((a + b - 1) / b); }

extern "C" void kernel_launch(void* const* d_in, const int* in_sizes, int n_in,
                              void* d_out, int out_size, void* d_ws, size_t ws_size,
                              hipStream_t stream)
{
    const float* x  = (const float*)d_in[0];
    const int*   ei = (const int*)  d_in[1];
    const float* W1 = (const float*)d_in[2];
    const float* b1 = (const float*)d_in[3];
    const float* W2 = (const float*)d_in[4];
    const float* b2 = (const float*)d_in[5];
    const float* Wc = (const float*)d_in[6];
    const float* bc = (const float*)d_in[7];
    const float* We = (const float*)d_in[8];
    const float* be = (const float*)d_in[9];
    float* out = (float*)d_out;

    const int* esrc = ei;
    const int* edst = ei + N_EDGES;

    // workspace carve-out (all offsets 16B aligned)
    float* ws = (float*)d_ws;
    float* deg    = ws; ws += N_NODES;
    float* dinv   = ws; ws += N_NODES;
    float* bufA   = ws; ws += (size_t)N_NODES * HID;   // GEMM output h@W
    float* bufB   = ws; ws += (size_t)N_NODES * HID;   // activations h
    float* bufC   = ws; ws += (size_t)N_NODES * HID;   // propagate / crf accum
    float* logits = ws; ws += (size_t)N_NODES * NCLS;
    float* ev     = ws; ws += (size_t)N_NODES * NCLS;
    float* hk     = ws; ws += (size_t)N_NODES * NCLS;
    float* tmpc   = ws; ws += (size_t)N_NODES * NCLS;

    const long nH  = (long)N_NODES * HID;
    const long nC  = (long)N_NODES * NCLS;
    const long nH4 = nH / 4;
    const long nC4 = nC / 4;

    // ---- gcn_norm: deg = in-degree + 1 (self loop); dinv = rsqrt(deg)
    k_fill<<<cdiv(N_NODES, 256), 256, 0, stream>>>(deg, 1.0f, N_NODES);
    k_deg_accum<<<cdiv(N_EDGES, 256), 256, 0, stream>>>(edst, deg, N_EDGES);
    k_dinv<<<cdiv(N_NODES, 256), 256, 0, stream>>>(deg, dinv, N_NODES);

    // ---- gcn1: h = relu(propagate(x@W1) + b1)
    dim3 g1(N_NODES / 16, HID / 64);                           // (3125, 2)
    k_gemm_wmma<FEAT, HID><<<g1, 128, 0, stream>>>(x, W1, nullptr, bufA, N_NODES, 0);
    k_init_self<HID><<<cdiv(nH, 256), 256, 0, stream>>>(bufA, dinv, bufC, N_NODES);
    k_scatter<HID / 4><<<cdiv((long)N_EDGES * (HID / 4), 256), 256, 0, stream>>>(
        esrc, edst, dinv, bufA, bufC, N_EDGES);
    k_bias_relu128<<<cdiv(nH4, 256), 256, 0, stream>>>(
        (const float4*)bufC, (const float4*)b1, (float4*)bufB, nH4);

    // ---- CRF smoothing: h = 0.9*h + 0.1*(scatter both directions)
    k_fill<<<cdiv(nH, 256), 256, 0, stream>>>(bufC, 0.0f, nH);
    k_crf_scatter<HID / 4><<<cdiv((long)N_EDGES * (HID / 4), 256), 256, 0, stream>>>(
        esrc, edst, bufB, bufC, N_EDGES);
    k_lerp<<<cdiv(nH4, 256), 256, 0, stream>>>(
        (const float4*)bufB, (const float4*)bufC, (float4*)bufB, 0.9f, 0.1f, nH4);

    // ---- gcn2: h2 = relu(propagate(h@W2) + b2)
    k_gemm_wmma<HID, HID><<<g1, 128, 0, stream>>>(bufB, W2, nullptr, bufA, N_NODES, 0);
    k_init_self<HID><<<cdiv(nH, 256), 256, 0, stream>>>(bufA, dinv, bufC, N_NODES);
    k_scatter<HID / 4><<<cdiv((long)N_EDGES * (HID / 4), 256), 256, 0, stream>>>(
        esrc, edst, dinv, bufA, bufC, N_EDGES);
    k_bias_relu128<<<cdiv(nH4, 256), 256, 0, stream>>>(
        (const float4*)bufC, (const float4*)b2, (float4*)bufB, nH4);

    // ---- classifier + evidence
    dim3 gc(N_NODES / 16, 1);
    k_gemm_wmma<HID, NCLS><<<gc, 128, 0, stream>>>(bufB, Wc, bc, logits, N_NODES, 0);
    k_gemm_wmma<NCLS, NCLS><<<gc, 128, 0, stream>>>(logits, We, be, ev, N_NODES, 1);

    // ---- APPNP: hk = 0.9*Ahat*hk + 0.1*ev, 10 rounds
    for (int k = 0; k < 10; ++k) {
        const float* cur = (k == 0) ? ev : hk;
        k_init_self<NCLS><<<cdiv(nC, 256), 256, 0, stream>>>(cur, dinv, tmpc, N_NODES);
        k_scatter<NCLS / 4><<<cdiv((long)N_EDGES * (NCLS / 4), 256), 256, 0, stream>>>(
            esrc, edst, dinv, cur, tmpc, N_EDGES);
        k_lerp<<<cdiv(nC4, 256), 256, 0, stream>>>(
            (const float4*)tmpc, (const float4*)ev, (float4*)hk, 0.9f, 0.1f, nC4);
    }

    // ---- log_softmax
    k_log_softmax<<<cdiv(N_NODES, 256), 256, 0, stream>>>(hk, out, N_NODES);
}